// MultiHeadAttentionPoolingLayer_51857435132415
// MI455X (gfx1250) — compile-verified
//
#include <hip/hip_runtime.h>
#include <math.h>

// MultiHeadAttentionPoolingLayer for MI455X (gfx1250, wave32).
//
// Bandwidth-bound problem (~540 MB traffic -> ~23us @ 23.3 TB/s). The score
// GEMV uses V_WMMA_F32_16X16X4_F32 (full fp32 precision, 16x16 tile, K=4,
// chained 32x for D=128). Everything else is streaming + segment reductions.
//
// vs round 1: W is staged into LDS pre-padded to 16 rows (rows H..15 = 0),
// removing the per-step v_pk_mul mask from the ds_load -> v_wmma chain.

#define D_DIM 128
#define H_DIM 4
#define NPAD 16                    // B-matrix columns (16x16 WMMA tile width)
#define TILE_NODES 64              // nodes per block in kernel 1 (4 waves x 16)
#define LDS_STRIDE 132             // 128 + 4 pad -> conflict-free b64 LDS reads
#define CHUNK 512                  // kernel-2 per-chunk node capacity in LDS

typedef __attribute__((ext_vector_type(2))) float v2f;
typedef __attribute__((ext_vector_type(8))) float v8f;

// ---------------------------------------------------------------------------
// Kernel 1: scores[n][h] = (x[n,:] . W[h,:] + b[h]) / temperature
// One wave computes a 16x16 WMMA tile (16 nodes x 16 cols, heads in cols 0..3).
// A (16x4 f32 per step):  lanes 0-15 hold M=lane, K = kb + {0,1};
//                         lanes 16-31 hold M=lane-16, K = kb + {2,3}.
// B (4x16 f32 per step):  mirrored layout, B[k][n] = W[n][k] (0 for n>=4,
//                         provided by the zero-padded LDS copy of W).
// C/D f32 16x16: VGPR r, lanes 0-15 -> (M=r, N=lane); lanes 16-31 -> (M=r+8).
// ---------------------------------------------------------------------------
__global__ __launch_bounds__(128)
void gap_scores_wmma(const float* __restrict__ x, const float* __restrict__ W,
                     const float* __restrict__ bvec, const float* __restrict__ temp,
                     float* __restrict__ scores, const int N)
{
    __shared__ __align__(16) float tile[TILE_NODES * LDS_STRIDE]; // ~33.8 KB
    __shared__ __align__(16) float wlds[NPAD * D_DIM];            // 8 KB

    const int tid   = threadIdx.x;
    const int node0 = blockIdx.x * TILE_NODES;

    // Stage 64x128 tile of x into LDS, coalesced float4 loads.
    for (int i = tid; i < TILE_NODES * (D_DIM / 4); i += 128) {
        const int r  = i >> 5;            // row in tile
        const int cq = (i & 31) << 2;     // column (float index)
        const int node = node0 + r;
        float4 v = make_float4(0.f, 0.f, 0.f, 0.f);
        if (node < N) v = *(const float4*)(x + (size_t)node * D_DIM + cq);
        *(float4*)(&tile[r * LDS_STRIDE + cq]) = v;
    }
    // Stage W zero-padded to 16 rows: rows 0..3 = W, rows 4..15 = 0.
    for (int i = tid; i < NPAD * D_DIM; i += 128)
        wlds[i] = (i < H_DIM * D_DIM) ? W[i] : 0.f;
    __syncthreads();

    const int wave = tid >> 5;
    const int lane = tid & 31;
    const int half = lane >> 4;           // 0 or 1 (selects K pair)
    const int l15  = lane & 15;

    const float* arow = &tile[(wave * 16 + l15) * LDS_STRIDE];
    const float* brow = &wlds[l15 * D_DIM];

    v8f c = {};
#pragma unroll
    for (int kb = 0; kb < D_DIM; kb += 4) {
        const int k2 = kb + (half << 1);
        const v2f a  = *(const v2f*)(arow + k2);   // ds_load_b64
        const v2f bm = *(const v2f*)(brow + k2);   // ds_load_b64 (pre-masked)
        // (neg_a, A, neg_b, B, c_mod, C, reuse_a, reuse_b)
        c = __builtin_amdgcn_wmma_f32_16x16x4_f32(false, a, false, bm,
                                                  (short)0, c, false, false);
    }

    // Lanes whose column is a real head write 8 node scores each.
    if (l15 < H_DIM) {
        const float invt = 1.0f / temp[0];
        const float bb   = bvec[l15];
        const int   nb   = node0 + wave * 16 + half * 8;
#pragma unroll
        for (int r = 0; r < 8; ++r) {
            const int node = nb + r;
            if (node < N)
                scores[(size_t)node * H_DIM + l15] = (c[r] + bb) * invt;
        }
    }
}

// ---------------------------------------------------------------------------
// Kernel 2: one block per segment. Segment bounds by binary search on the
// sorted batch_indices. Per-head max + exp-sum reductions (wave32 shuffles +
// LDS), attention write-out in [H][N] layout, then pooled[b][d] with one
// feature lane per thread (coalesced reads of x).
// ---------------------------------------------------------------------------
__device__ __forceinline__ float4 max4(float4 a, float4 b) {
    return make_float4(fmaxf(a.x, b.x), fmaxf(a.y, b.y),
                       fmaxf(a.z, b.z), fmaxf(a.w, b.w));
}

__global__ __launch_bounds__(128)
void gap_segment_pool(const float* __restrict__ x, const int* __restrict__ ids,
                      const float* __restrict__ scores,
                      float* __restrict__ pooled, float* __restrict__ attn,
                      const int N)
{
    const int b    = blockIdx.x;
    const int tid  = threadIdx.x;       // 128 threads: tid == feature lane d
    const int lane = tid & 31;
    const int wave = tid >> 5;

    // lower_bound(ids, b) and lower_bound(ids, b+1)
    int lo = 0, hi = N;
    while (lo < hi) { const int mid = (lo + hi) >> 1;
                      if (ids[mid] <  b) lo = mid + 1; else hi = mid; }
    const int start = lo;
    hi = N;
    while (lo < hi) { const int mid = (lo + hi) >> 1;
                      if (ids[mid] <= b) lo = mid + 1; else hi = mid; }
    const int end = lo;

    __shared__ float4 red[4];
    __shared__ float  cw[CHUNK];

    const float ninf = -__builtin_inff();

    // ---- per-head max over the segment ----
    float4 mx = make_float4(ninf, ninf, ninf, ninf);
    for (int n = start + tid; n < end; n += 128)
        mx = max4(mx, *(const float4*)(scores + (size_t)n * H_DIM));
#pragma unroll
    for (int off = 16; off > 0; off >>= 1) {
        mx.x = fmaxf(mx.x, __shfl_xor(mx.x, off));
        mx.y = fmaxf(mx.y, __shfl_xor(mx.y, off));
        mx.z = fmaxf(mx.z, __shfl_xor(mx.z, off));
        mx.w = fmaxf(mx.w, __shfl_xor(mx.w, off));
    }
    if (lane == 0) red[wave] = mx;
    __syncthreads();
    const float4 m = max4(max4(red[0], red[1]), max4(red[2], red[3]));
    __syncthreads();

    // ---- per-head sum of exp(score - max) ----
    float4 sm = make_float4(0.f, 0.f, 0.f, 0.f);
    for (int n = start + tid; n < end; n += 128) {
        const float4 s = *(const float4*)(scores + (size_t)n * H_DIM);
        sm.x += __expf(s.x - m.x);
        sm.y += __expf(s.y - m.y);
        sm.z += __expf(s.z - m.z);
        sm.w += __expf(s.w - m.w);
    }
#pragma unroll
    for (int off = 16; off > 0; off >>= 1) {
        sm.x += __shfl_xor(sm.x, off);
        sm.y += __shfl_xor(sm.y, off);
        sm.z += __shfl_xor(sm.z, off);
        sm.w += __shfl_xor(sm.w, off);
    }
    if (lane == 0) red[wave] = sm;
    __syncthreads();
    float4 tot;
    tot.x = red[0].x + red[1].x + red[2].x + red[3].x;
    tot.y = red[0].y + red[1].y + red[2].y + red[3].y;
    tot.z = red[0].z + red[1].z + red[2].z + red[3].z;
    tot.w = red[0].w + red[1].w + red[2].w + red[3].w;
    float4 inv;
    inv.x = (tot.x > 0.f) ? 1.f / tot.x : 0.f;
    inv.y = (tot.y > 0.f) ? 1.f / tot.y : 0.f;
    inv.z = (tot.z > 0.f) ? 1.f / tot.z : 0.f;
    inv.w = (tot.w > 0.f) ? 1.f / tot.w : 0.f;
    __syncthreads();

    // ---- attn write-out + weighted pooling (chunked through LDS) ----
    float acc = 0.f;
    for (int cs = start; cs < end; cs += CHUNK) {
        const int ce = (cs + CHUNK < end) ? (cs + CHUNK) : end;
        for (int n = cs + tid; n < ce; n += 128) {
            const float4 s = *(const float4*)(scores + (size_t)n * H_DIM);
            const float a0 = __expf(s.x - m.x) * inv.x;
            const float a1 = __expf(s.y - m.y) * inv.y;
            const float a2 = __expf(s.z - m.z) * inv.z;
            const float a3 = __expf(s.w - m.w) * inv.w;
            attn[(size_t)0 * N + n] = a0;
            attn[(size_t)1 * N + n] = a1;
            attn[(size_t)2 * N + n] = a2;
            attn[(size_t)3 * N + n] = a3;
            cw[n - cs] = 0.25f * (a0 + a1 + a2 + a3);  // mean over H=4 heads
        }
        __syncthreads();
        for (int n = cs; n < ce; ++n)
            acc = fmaf(x[(size_t)n * D_DIM + tid], cw[n - cs], acc);
        __syncthreads();
    }
    pooled[(size_t)b * D_DIM + tid] = acc;
}

// ---------------------------------------------------------------------------
// Launcher. Inputs: x[N*D] f32, batch_indices[N] i32, W[H*D] f32, b[H] f32,
// temperature[1] f32. Output: pooled[B*D] then attn[H*N], concatenated f32.
// Workspace: scores[N*H] f32 (8 MB).
// ---------------------------------------------------------------------------
extern "C" void kernel_launch(void* const* d_in, const int* in_sizes, int n_in,
                              void* d_out, int out_size, void* d_ws, size_t ws_size,
                              hipStream_t stream)
{
    const float* x    = (const float*)d_in[0];
    const int*   ids  = (const int*)d_in[1];
    const float* W    = (const float*)d_in[2];
    const float* bvec = (const float*)d_in[3];
    const float* temp = (const float*)d_in[4];

    const int N = in_sizes[1];
    const int B = (out_size - H_DIM * N) / D_DIM;

    float* pooled = (float*)d_out;
    float* attn   = (float*)d_out + (size_t)B * D_DIM;
    float* scores = (float*)d_ws;

    const int blocks1 = (N + TILE_NODES - 1) / TILE_NODES;
    gap_scores_wmma<<<blocks1, 128, 0, stream>>>(x, W, bvec, temp, scores, N);
    gap_segment_pool<<<B, 128, 0, stream>>>(x, ids, scores, pooled, attn, N);
}